// GIN_83348135346296
// MI455X (gfx1250) — compile-verified
//
#include <hip/hip_runtime.h>
#include <math.h>

#define NN 50000
#define NE 800000
#define EMB 64

typedef float v2f __attribute__((ext_vector_type(2)));
typedef float v8f __attribute__((ext_vector_type(8)));

static __device__ __forceinline__ float leaky02(float x) {
    return x >= 0.0f ? x : 0.2f * x;
}

// float atomic-max via monotone int encoding; buffer must be initialized to -inf.
static __device__ __forceinline__ void atomicMaxF32(float* addr, float val) {
    if (!(val < 0.0f)) {
        atomicMax((int*)addr, __float_as_int(val));
    } else {
        atomicMin((unsigned int*)addr, __float_as_uint(val));
    }
}

// -------------------- embedding gather: h0[n,:] = emb[feat[n],:] * fval[n] --------------------
__global__ void embed_kernel(const int* __restrict__ feats, const float* __restrict__ fvals,
                             const float* __restrict__ emb, float* __restrict__ h0) {
    int t = blockIdx.x * blockDim.x + threadIdx.x;        // NN*16 threads, float4 each
    if (t >= NN * 16) return;
    int n = t >> 4;
    int c = (t & 15) << 2;
    int row = feats[n];
    float s = fvals[n];
    const float4 v = *(const float4*)(emb + (size_t)row * EMB + c);
    float4 o;
    o.x = v.x * s; o.y = v.y * s; o.z = v.z * s; o.w = v.w * s;
    *(float4*)(h0 + (size_t)n * EMB + c) = o;
}

// -------------------- f32 WMMA GEMM: C[MxNc] = A[MxK] @ B[KxNc] (+ bias[col]) ----------------
// one wave per 16x16 output tile; grid = (M/16, Nc/16); block = 32 (wave32, EXEC all ones).
// K, Nc compile-time so all strides fold into immediate load offsets; K-loop fully unrolled
// so the WMMA chain streams through the XDL pipe with clause-grouped loads.
template <int K, int Nc>
__global__ void __launch_bounds__(32)
wmma_gemm_f32(const float* __restrict__ A, const float* __restrict__ B,
              float* __restrict__ C, const float* __restrict__ bias) {
    const int lane = threadIdx.x;                 // 0..31
    const int row0 = blockIdx.x << 4;
    const int col0 = blockIdx.y << 4;
    const int r    = lane & 15;
    const int kb   = (lane >> 4) << 1;            // 0 or 2 (K sub-pair per half-wave)

    const float* __restrict__ Arow = A + (size_t)(row0 + r) * K + kb;
    const float* __restrict__ Bcol = B + (size_t)kb * Nc + col0 + r;

    v8f acc = {};
#pragma unroll
    for (int k0 = 0; k0 < K; k0 += 4) {
        v2f a, b;
        a.x = Arow[k0];                           // A: 16x4 tile, per ISA layout
        a.y = Arow[k0 + 1];
        b.x = Bcol[k0 * Nc];                      // B: 4x16 tile, row-striped across lanes
        b.y = Bcol[(k0 + 1) * Nc];
        acc = __builtin_amdgcn_wmma_f32_16x16x4_f32(
            /*neg_a=*/false, a, /*neg_b=*/false, b,
            /*c_mod=*/(short)0, acc, /*reuse_a=*/false, /*reuse_b=*/false);
    }
    const float bv = bias ? bias[col0 + r] : 0.0f;
    const int rbase = row0 + ((lane >> 4) << 3);  // D: VGPR i = row i (+8 for upper half-wave)
#pragma unroll
    for (int i = 0; i < 8; ++i) {
        C[(size_t)(rbase + i) * Nc + col0 + r] = acc[i] + bv;
    }
}

// -------------------- attention scores: el/er[n,h] = <fh[n,h,:], al/ar[h,:]> -----------------
template <int H>
__global__ void attn_scores_kernel(const float* __restrict__ fh,
                                   const float* __restrict__ al, const float* __restrict__ ar,
                                   float* __restrict__ el, float* __restrict__ er) {
    int t = blockIdx.x * blockDim.x + threadIdx.x;
    if (t >= NN * H) return;
    int n = (H == 1) ? t : (t >> 1);
    int h = (H == 1) ? 0 : (t & 1);
    const float* __restrict__ row = fh + (size_t)n * (H * 64) + h * 64;
    float sl = 0.0f, sr = 0.0f;
#pragma unroll 8
    for (int d = 0; d < 64; ++d) {
        float v = row[d];
        sl = fmaf(v, al[h * 64 + d], sl);
        sr = fmaf(v, ar[h * 64 + d], sr);
    }
    el[t] = sl;
    er[t] = sr;
}

// -------------------- out[n,c] = bias[c] (+ resid[n,c]) : accumulator base -------------------
__global__ void init_out_kernel(float* __restrict__ out, const float* __restrict__ bias,
                                const float* __restrict__ resid, int C) {
    int t = blockIdx.x * blockDim.x + threadIdx.x;
    if (t >= NN * C) return;
    int c = t & (C - 1);                           // C is 128 (power of two)
    float v = bias[c];
    if (resid) v += resid[t];
    out[t] = v;
}

// -------------------- mx = -inf, dn = 0 ------------------------------------------------------
__global__ void init_softmax_kernel(float* __restrict__ mx, float* __restrict__ dn, int cnt) {
    int t = blockIdx.x * blockDim.x + threadIdx.x;
    if (t >= cnt) return;
    mx[t] = -INFINITY;
    dn[t] = 0.0f;
}

// -------------------- pass 1: segment max over edges -----------------------------------------
template <int H>
__global__ void edge_max_kernel(const int* __restrict__ src, const int* __restrict__ dst,
                                const float* __restrict__ el, const float* __restrict__ er,
                                float* __restrict__ mx) {
    int t = blockIdx.x * blockDim.x + threadIdx.x;
    if (t >= NE * H) return;
    int e = (H == 1) ? t : (t >> 1);
    int h = (H == 1) ? 0 : (t & 1);
    int s = src[e], d = dst[e];
    float v = leaky02(el[(size_t)s * H + h] + er[(size_t)d * H + h]);
    atomicMaxF32(&mx[(size_t)d * H + h], v);
}

// -------------------- m = isfinite(m) ? m : 0 -------------------------------------------------
__global__ void sanitize_max_kernel(float* __restrict__ mx, int cnt) {
    int t = blockIdx.x * blockDim.x + threadIdx.x;
    if (t >= cnt) return;
    float m = mx[t];
    mx[t] = (m == -INFINITY) ? 0.0f : m;
}

// -------------------- pass 2: ex = exp(e - m[dst]); den += ex ---------------------------------
template <int H>
__global__ void edge_exp_kernel(const int* __restrict__ src, const int* __restrict__ dst,
                                const float* __restrict__ el, const float* __restrict__ er,
                                const float* __restrict__ mx, float* __restrict__ exb,
                                float* __restrict__ dn) {
    int t = blockIdx.x * blockDim.x + threadIdx.x;
    if (t >= NE * H) return;
    int e = (H == 1) ? t : (t >> 1);
    int h = (H == 1) ? 0 : (t & 1);
    int s = src[e], d = dst[e];
    float v  = leaky02(el[(size_t)s * H + h] + er[(size_t)d * H + h]);
    float ex = expf(v - mx[(size_t)d * H + h]);
    exb[t] = ex;
    atomicAdd(&dn[(size_t)d * H + h], ex);
}

// -------------------- pass 3: out[dst,h,:] += alpha * fh[src,h,:] (float4 + f32 atomics) -----
template <int H>
__global__ void edge_aggr_kernel(const int* __restrict__ src, const int* __restrict__ dst,
                                 const float* __restrict__ exb, const float* __restrict__ dn,
                                 const float* __restrict__ fh, float* __restrict__ out) {
    constexpr int CPE   = H * 16;                  // float4 chunks per edge (32 or 16)
    constexpr int SHIFT = (H == 2) ? 5 : 4;
    long long t = (long long)blockIdx.x * blockDim.x + threadIdx.x;
    if (t >= (long long)NE * CPE) return;
    int e = (int)(t >> SHIFT);
    int c = (int)t & (CPE - 1);
    int head = c >> 4;
    int s = src[e], d = dst[e];
    float alpha = exb[(size_t)e * H + head] / fmaxf(dn[(size_t)d * H + head], 1e-9f);
    constexpr int ROWW = H * 64;
    const float4 v = *(const float4*)(fh + (size_t)s * ROWW + (c << 2));
    float* o = out + (size_t)d * ROWW + (c << 2);
    atomicAdd(o + 0, alpha * v.x);
    atomicAdd(o + 1, alpha * v.y);
    atomicAdd(o + 2, alpha * v.z);
    atomicAdd(o + 3, alpha * v.w);
}

// -------------------- ELU in place ------------------------------------------------------------
__global__ void elu_kernel(float* __restrict__ x, int cnt) {
    int t = blockIdx.x * blockDim.x + threadIdx.x;
    if (t >= cnt) return;
    float v = x[t];
    x[t] = (v > 0.0f) ? v : expm1f(v);
}

static inline int cdiv(long long a, int b) { return (int)((a + b - 1) / b); }

extern "C" void kernel_launch(void* const* d_in, const int* in_sizes, int n_in,
                              void* d_out, int out_size, void* d_ws, size_t ws_size,
                              hipStream_t stream) {
    const int*   feats = (const int*)d_in[0];      // (8, NN) - row 0 used
    const float* fvals = (const float*)d_in[1];    // (8, NN) - row 0 used
    const int*   src   = (const int*)d_in[2];
    const int*   dst   = (const int*)d_in[3];
    const float* emb   = (const float*)d_in[4];
    const float* W0    = (const float*)d_in[5];
    const float* al0   = (const float*)d_in[6];
    const float* ar0   = (const float*)d_in[7];
    const float* b0    = (const float*)d_in[8];
    const float* W1    = (const float*)d_in[9];
    const float* al1   = (const float*)d_in[10];
    const float* ar1   = (const float*)d_in[11];
    const float* b1    = (const float*)d_in[12];
    const float* W2    = (const float*)d_in[13];
    const float* al2   = (const float*)d_in[14];
    const float* ar2   = (const float*)d_in[15];
    const float* b2    = (const float*)d_in[16];
    const float* resW2 = (const float*)d_in[17];
    float* out = (float*)d_out;                    // (NN, 64)

    // workspace carve-up (floats)
    float* ws  = (float*)d_ws;
    float* hA  = ws;                               // NN*128
    float* hB  = hA + (size_t)NN * 128;            // NN*128 (layer-0 input h0 lives here, ld=64)
    float* fh  = hB + (size_t)NN * 128;            // NN*128
    float* exb = fh + (size_t)NN * 128;            // NE*2
    float* el  = exb + (size_t)NE * 2;             // NN*2
    float* er  = el + (size_t)NN * 2;
    float* mx  = er + (size_t)NN * 2;
    float* dn  = mx + (size_t)NN * 2;

    const int T = 256;

    // ---- embedding: h0 (= hB, ld 64) ----
    embed_kernel<<<cdiv((long long)NN * 16, T), T, 0, stream>>>(feats, fvals, emb, hB);

    // ================= layer 0: in=64, H=2, D=64, no residual, ELU =================
    {
        dim3 g(NN / 16, 128 / 16);
        wmma_gemm_f32<64, 128><<<g, 32, 0, stream>>>(hB, W0, fh, nullptr);
        attn_scores_kernel<2><<<cdiv((long long)NN * 2, T), T, 0, stream>>>(fh, al0, ar0, el, er);
        init_out_kernel<<<cdiv((long long)NN * 128, T), T, 0, stream>>>(hA, b0, nullptr, 128);
        init_softmax_kernel<<<cdiv((long long)NN * 2, T), T, 0, stream>>>(mx, dn, NN * 2);
        edge_max_kernel<2><<<cdiv((long long)NE * 2, T), T, 0, stream>>>(src, dst, el, er, mx);
        sanitize_max_kernel<<<cdiv((long long)NN * 2, T), T, 0, stream>>>(mx, NN * 2);
        edge_exp_kernel<2><<<cdiv((long long)NE * 2, T), T, 0, stream>>>(src, dst, el, er, mx, exb, dn);
        edge_aggr_kernel<2><<<cdiv((long long)NE * 32, T), T, 0, stream>>>(src, dst, exb, dn, fh, hA);
        elu_kernel<<<cdiv((long long)NN * 128, T), T, 0, stream>>>(hA, NN * 128);
    }

    // ================= layer 1: in=128, H=2, D=64, identity residual, ELU ==========
    {
        dim3 g(NN / 16, 128 / 16);
        wmma_gemm_f32<128, 128><<<g, 32, 0, stream>>>(hA, W1, fh, nullptr);
        attn_scores_kernel<2><<<cdiv((long long)NN * 2, T), T, 0, stream>>>(fh, al1, ar1, el, er);
        init_out_kernel<<<cdiv((long long)NN * 128, T), T, 0, stream>>>(hB, b1, hA, 128);
        init_softmax_kernel<<<cdiv((long long)NN * 2, T), T, 0, stream>>>(mx, dn, NN * 2);
        edge_max_kernel<2><<<cdiv((long long)NE * 2, T), T, 0, stream>>>(src, dst, el, er, mx);
        sanitize_max_kernel<<<cdiv((long long)NN * 2, T), T, 0, stream>>>(mx, NN * 2);
        edge_exp_kernel<2><<<cdiv((long long)NE * 2, T), T, 0, stream>>>(src, dst, el, er, mx, exb, dn);
        edge_aggr_kernel<2><<<cdiv((long long)NE * 32, T), T, 0, stream>>>(src, dst, exb, dn, fh, hB);
        elu_kernel<<<cdiv((long long)NN * 128, T), T, 0, stream>>>(hB, NN * 128);
    }

    // ================= layer 2: in=128, H=1, D=64, projected residual, no act ======
    {
        dim3 g(NN / 16, 64 / 16);
        wmma_gemm_f32<128, 64><<<g, 32, 0, stream>>>(hB, W2, fh, nullptr);       // fh (ld 64)
        attn_scores_kernel<1><<<cdiv((long long)NN * 1, T), T, 0, stream>>>(fh, al2, ar2, el, er);
        wmma_gemm_f32<128, 64><<<g, 32, 0, stream>>>(hB, resW2, out, b2);        // out = h@resW2 + b2
        init_softmax_kernel<<<cdiv((long long)NN * 1, T), T, 0, stream>>>(mx, dn, NN * 1);
        edge_max_kernel<1><<<cdiv((long long)NE * 1, T), T, 0, stream>>>(src, dst, el, er, mx);
        sanitize_max_kernel<<<cdiv((long long)NN * 1, T), T, 0, stream>>>(mx, NN * 1);
        edge_exp_kernel<1><<<cdiv((long long)NE * 1, T), T, 0, stream>>>(src, dst, el, er, mx, exb, dn);
        edge_aggr_kernel<1><<<cdiv((long long)NE * 16, T), T, 0, stream>>>(src, dst, exb, dn, fh, out);
        // H=1: mean over heads is identity; d_out already final
    }
}